// VAE_24352464569663
// MI455X (gfx1250) — compile-verified
//
#include <hip/hip_runtime.h>
#include <math.h>

typedef __attribute__((ext_vector_type(2))) float v2f;
typedef __attribute__((ext_vector_type(8))) float v8f;

// ---------------- output layout (floats), return order: rec, cov, sigma, z, Dinv ----------------
#define REC_OFF   0         // 16*1*50*50      = 40000
#define COV_OFF   40000     // 16*64*64        = 65536
#define SIG_OFF   105536    // 64*64           = 4096
#define Z_OFF     109632    // 16*64*64        = 65536
#define DINV_OFF  175168    // 64*64 complex64 = 8192 floats (interleaved re,im)

// ---------------- workspace layout (floats) ----------------
#define WS_DNR   0                   // 4096  normalized DFT real
#define WS_DNI   4096                // 4096  normalized DFT imag
#define WS_DVR   8192                // 4096  Dinv real
#define WS_DVI   12288               // 4096  Dinv imag
#define WS_Z     16384               // 65536 z flat (B,64,64)
#define WS_Z1    (16384 + 65536)     // 3200  decoder input
#define WS_Y2    (WS_Z1 + 3200)      // 30976  (16,16,11,11)
#define WS_Y3    (WS_Y2 + 30976)    // 270848 (16,32,23,23)
#define WS_Y4    (WS_Y3 + 270848)   // 640000 (16,64,25,25)

// =====================================================================================
// Kernel 1: build normalized DFT matrix, Dinv, sigma_matrix, and emit Dinv to d_out.
// D_raw[j][k] = round(cos(2*pi*j*k/64)) + i*round(sin(...)); row-L2-normalize -> DN.
// Dinv = rownorm(conj(DN^T)).  One block, 64 threads (thread j = row j).
// =====================================================================================
__global__ void dft_sigma_kernel(float* __restrict__ ws, float* __restrict__ out,
                                 const float* __restrict__ x) {
  __shared__ float sdr[64 * 64];   // 16 KB, normalized DN real
  __shared__ float sdi[64 * 64];   // 16 KB, normalized DN imag
  const int j = threadIdx.x;       // 0..63
  const float TWO_PI = 6.28318530717958647692f;

  // pass 1: row norm of raw rounded DFT row j
  float nrm = 0.f;
  for (int k = 0; k < 64; ++k) {
    float ang = TWO_PI * (float)((j * k) & 63) * (1.0f / 64.0f);
    float cr = roundf(cosf(ang));
    float ci = roundf(sinf(ang));
    nrm += cr * cr + ci * ci;
  }
  float inv = 1.0f / fmaxf(sqrtf(nrm), 1e-12f);
  // pass 2: store normalized row
  for (int k = 0; k < 64; ++k) {
    float ang = TWO_PI * (float)((j * k) & 63) * (1.0f / 64.0f);
    float cr = roundf(cosf(ang)) * inv;
    float ci = roundf(sinf(ang)) * inv;
    sdr[j * 64 + k] = cr;
    sdi[j * 64 + k] = ci;
    ws[WS_DNR + j * 64 + k] = cr;
    ws[WS_DNI + j * 64 + k] = ci;
  }
  __syncthreads();

  // Dinv row j = conj(DN column j) / ||DN column j||
  float cn = 0.f;
  for (int k = 0; k < 64; ++k) {
    float a = sdr[k * 64 + j], b = sdi[k * 64 + j];
    cn += a * a + b * b;
  }
  float cinv = 1.0f / fmaxf(sqrtf(cn), 1e-12f);
  for (int k = 0; k < 64; ++k) {
    float dr = sdr[k * 64 + j] * cinv;
    float di = -sdi[k * 64 + j] * cinv;
    ws[WS_DVR + j * 64 + k] = dr;
    ws[WS_DVI + j * 64 + k] = di;
    out[DINV_OFF + 2 * (j * 64 + k) + 0] = dr;   // complex64 interleaved
    out[DINV_OFF + 2 * (j * 64 + k) + 1] = di;
  }

  // sigma_matrix = diag(x[B-1])
  float xv = x[15 * 64 + j];
  for (int l = 0; l < 64; ++l)
    out[SIG_OFF + j * 64 + l] = (l == j) ? xv : 0.0f;
}

// =====================================================================================
// Kernel 2: per-batch sampling using V_WMMA_F32_16X16X4_F32.
//   Stage 1: cov[b] = (DNr .* x) @ DinvR + (-DNi .* x) @ DinvI      (col-scaled by x[b,k])
//   Stage 2: z[b]   = A1 @ eps_b + A2 @ eps_a, with sr/si from sign(x).
// One block per batch (16 blocks), 256 threads = 8 wave32s, 64 KB LDS reused across stages.
// Each wave computes two 16x16 output tiles per stage: 32 WMMAs accumulated per tile.
// =====================================================================================
__global__ void sampling_kernel(float* __restrict__ ws, float* __restrict__ out,
                                const float* __restrict__ x,
                                const float* __restrict__ eps_a,
                                const float* __restrict__ eps_b) {
  __shared__ float sP0[64 * 64];  // A-operand matrix 0
  __shared__ float sP1[64 * 64];  // A-operand matrix 1
  __shared__ float sQ0[64 * 64];  // B-operand matrix 0
  __shared__ float sQ1[64 * 64];  // B-operand matrix 1

  const int b = blockIdx.x;
  const int t = threadIdx.x;
  const int lane = t & 31;
  const int wave = t >> 5;
  const int m    = lane & 15;    // row within tile (A), col within tile (B/C/D)
  const int half = lane >> 4;    // 0: K base +0, 1: K base +2

  // ---------------- stage 1: cov ----------------
  for (int i = t; i < 4096; i += 256) {
    int k = i & 63;
    float xv = x[b * 64 + k];
    float dr = ws[WS_DNR + i], di = ws[WS_DNI + i];
    sP0[i] = dr * xv;       // Ar
    sP1[i] = -di * xv;      // -Ai (so we can accumulate with plain adds)
    sQ0[i] = ws[WS_DVR + i];
    sQ1[i] = ws[WS_DVI + i];
  }
  __syncthreads();

  for (int tt = 0; tt < 2; ++tt) {
    int tile = wave * 2 + tt;        // 0..15
    int jt = tile >> 2, lt = tile & 3;
    v8f acc = {0.f, 0.f, 0.f, 0.f, 0.f, 0.f, 0.f, 0.f};
    for (int kk = 0; kk < 16; ++kk) {
      int k0 = 4 * kk + 2 * half;
      v2f a0, b0, a1, b1;
      a0.x = sP0[(jt * 16 + m) * 64 + k0];
      a0.y = sP0[(jt * 16 + m) * 64 + k0 + 1];
      b0.x = sQ0[(k0) * 64 + lt * 16 + m];
      b0.y = sQ0[(k0 + 1) * 64 + lt * 16 + m];
      acc = __builtin_amdgcn_wmma_f32_16x16x4_f32(false, a0, false, b0,
                                                  (short)0, acc, false, false);
      a1.x = sP1[(jt * 16 + m) * 64 + k0];
      a1.y = sP1[(jt * 16 + m) * 64 + k0 + 1];
      b1.x = sQ1[(k0) * 64 + lt * 16 + m];
      b1.y = sQ1[(k0 + 1) * 64 + lt * 16 + m];
      acc = __builtin_amdgcn_wmma_f32_16x16x4_f32(false, a1, false, b1,
                                                  (short)0, acc, false, false);
    }
    for (int v = 0; v < 8; ++v) {
      int row = jt * 16 + (half ? (v + 8) : v);
      int col = lt * 16 + m;
      out[COV_OFF + b * 4096 + row * 64 + col] = acc[v];
    }
  }
  __syncthreads();   // all tile reads done before LDS reuse

  // ---------------- stage 2: z ----------------
  for (int i = t; i < 4096; i += 256) {
    int k = i & 63;
    float xv = x[b * 64 + k];
    float sr = (xv > 0.f) ? 1.f : 0.f;     // sqrt(x)/|sqrt(x)| = 1 for x>0
    float si = (xv < 0.f) ? 1.f : 0.f;     //                   = i for x<0
    float dr = ws[WS_DNR + i], di = ws[WS_DNI + i];
    sP0[i] = dr * sr - di * si;            // A1 (real part of q)
    sP1[i] = dr * si + di * sr;            // A2 (imag part of q)
    sQ0[i] = eps_b[b * 4096 + i];
    sQ1[i] = eps_a[b * 4096 + i];
  }
  __syncthreads();

  for (int tt = 0; tt < 2; ++tt) {
    int tile = wave * 2 + tt;
    int jt = tile >> 2, lt = tile & 3;
    v8f acc = {0.f, 0.f, 0.f, 0.f, 0.f, 0.f, 0.f, 0.f};
    for (int kk = 0; kk < 16; ++kk) {
      int k0 = 4 * kk + 2 * half;
      v2f a0, b0, a1, b1;
      a0.x = sP0[(jt * 16 + m) * 64 + k0];
      a0.y = sP0[(jt * 16 + m) * 64 + k0 + 1];
      b0.x = sQ0[(k0) * 64 + lt * 16 + m];
      b0.y = sQ0[(k0 + 1) * 64 + lt * 16 + m];
      acc = __builtin_amdgcn_wmma_f32_16x16x4_f32(false, a0, false, b0,
                                                  (short)0, acc, false, false);
      a1.x = sP1[(jt * 16 + m) * 64 + k0];
      a1.y = sP1[(jt * 16 + m) * 64 + k0 + 1];
      b1.x = sQ1[(k0) * 64 + lt * 16 + m];
      b1.y = sQ1[(k0 + 1) * 64 + lt * 16 + m];
      acc = __builtin_amdgcn_wmma_f32_16x16x4_f32(false, a1, false, b1,
                                                  (short)0, acc, false, false);
    }
    for (int v = 0; v < 8; ++v) {
      int row = jt * 16 + (half ? (v + 8) : v);
      int col = lt * 16 + m;
      float zv = acc[v];
      out[Z_OFF + b * 4096 + row * 64 + col] = zv;
      ws[WS_Z + b * 4096 + row * 64 + col]   = zv;
    }
  }
}

// =====================================================================================
// Kernel 3: z1 = W_lin @ z.flat + b_lin.  (3200 x 65536) GEMV — pure HBM bandwidth:
// 838 MB of weights streamed once with float4 loads. One block per output row.
// =====================================================================================
__global__ void gemv_kernel(const float* __restrict__ W, const float* __restrict__ bias,
                            const float* __restrict__ ws, float* __restrict__ z1) {
  const int r = blockIdx.x;
  const int t = threadIdx.x;     // 256
  const float4* __restrict__ Wr = (const float4*)(W + (size_t)r * 65536);
  const float4* __restrict__ zv = (const float4*)(ws + WS_Z);
  float acc = 0.f;
#pragma unroll 4
  for (int i = t; i < 16384; i += 256) {
    float4 w4 = Wr[i];
    float4 z4 = zv[i];
    acc += w4.x * z4.x + w4.y * z4.y + w4.z * z4.z + w4.w * z4.w;
  }
  __shared__ float red[256];
  red[t] = acc;
  __syncthreads();
  for (int s = 128; s > 0; s >>= 1) {
    if (t < s) red[t] += red[t + s];
    __syncthreads();
  }
  if (t == 0) z1[r] = red[0] + bias[r];
}

// =====================================================================================
// Kernel 4: transposed convolution (gather form), torch ConvTranspose2d semantics,
// weight layout (Cin, Cout, K, K).  act: 0 = LeakyReLU(0.01), 1 = sigmoid.
// =====================================================================================
__global__ void convt_kernel(const float* __restrict__ in, const float* __restrict__ w,
                             const float* __restrict__ bias, float* __restrict__ out,
                             int Cin, int Cout, int H, int Wd, int K, int S,
                             int OH, int OW, int act, int total) {
  int idx = blockIdx.x * blockDim.x + threadIdx.x;
  if (idx >= total) return;
  int ox = idx % OW;
  int oy = (idx / OW) % OH;
  int co = (idx / (OW * OH)) % Cout;
  int b  = idx / (OW * OH * Cout);
  float acc = bias[co];
  for (int ky = 0; ky < K; ++ky) {
    int ty = oy - ky;
    if (ty < 0 || (ty % S) != 0) continue;
    int iy = ty / S;
    if (iy >= H) continue;
    for (int kx = 0; kx < K; ++kx) {
      int tx = ox - kx;
      if (tx < 0 || (tx % S) != 0) continue;
      int ix = tx / S;
      if (ix >= Wd) continue;
      const float* ip = in + ((b * Cin) * H + iy) * Wd + ix;
      const float* wp = w + (co * K + ky) * K + kx;
      for (int ci = 0; ci < Cin; ++ci) {
        acc += wp[ci * Cout * K * K] * ip[ci * H * Wd];
      }
    }
  }
  if (act == 0) acc = (acc >= 0.f) ? acc : 0.01f * acc;
  else          acc = 1.0f / (1.0f + expf(-acc));
  out[idx] = acc;
}

// =====================================================================================
extern "C" void kernel_launch(void* const* d_in, const int* in_sizes, int n_in,
                              void* d_out, int out_size, void* d_ws, size_t ws_size,
                              hipStream_t stream) {
  const float* x     = (const float*)d_in[0];
  const float* eps_a = (const float*)d_in[1];
  const float* eps_b = (const float*)d_in[2];
  const float* W_lin = (const float*)d_in[3];
  const float* b_lin = (const float*)d_in[4];
  const float* w2 = (const float*)d_in[5];
  const float* b2 = (const float*)d_in[6];
  const float* w3 = (const float*)d_in[7];
  const float* b3 = (const float*)d_in[8];
  const float* w4 = (const float*)d_in[9];
  const float* b4 = (const float*)d_in[10];
  const float* w5 = (const float*)d_in[11];
  const float* b5 = (const float*)d_in[12];

  float* out = (float*)d_out;
  float* ws  = (float*)d_ws;

  // constants + sigma + Dinv output
  dft_sigma_kernel<<<1, 64, 0, stream>>>(ws, out, x);
  // cov + z via WMMA f32 16x16x4
  sampling_kernel<<<16, 256, 0, stream>>>(ws, out, x, eps_a, eps_b);
  // bandwidth-dominant GEMV
  gemv_kernel<<<3200, 256, 0, stream>>>(W_lin, b_lin, ws, ws + WS_Z1);
  // decoder convT chain
  convt_kernel<<<(30976 + 255) / 256, 256, 0, stream>>>(
      ws + WS_Z1, w2, b2, ws + WS_Y2, 8, 16, 5, 5, 3, 2, 11, 11, 0, 30976);
  convt_kernel<<<(270848 + 255) / 256, 256, 0, stream>>>(
      ws + WS_Y2, w3, b3, ws + WS_Y3, 16, 32, 11, 11, 3, 2, 23, 23, 0, 270848);
  convt_kernel<<<(640000 + 255) / 256, 256, 0, stream>>>(
      ws + WS_Y3, w4, b4, ws + WS_Y4, 32, 64, 23, 23, 3, 1, 25, 25, 0, 640000);
  convt_kernel<<<(40000 + 255) / 256, 256, 0, stream>>>(
      ws + WS_Y4, w5, b5, out + REC_OFF, 64, 1, 25, 25, 2, 2, 50, 50, 1, 40000);
}